// LightGCN_36163624632738
// MI455X (gfx1250) — compile-verified
//
#include <hip/hip_runtime.h>
#include <hip/hip_bf16.h>

#define N_USERS   100000
#define N_ITEMS   50000
#define N_NODES   (N_USERS + N_ITEMS)     // 150000
#define LATENT    64
#define NUM_LAYERS 3
#define NELEMS    (N_NODES * LATENT)      // 9,600,000 floats (38.4 MB)

typedef __attribute__((ext_vector_type(2))) float v2f;
typedef __attribute__((ext_vector_type(8))) float v8f;

// ---------------------------------------------------------------------------
// init: h = ego = concat(user_emb, item_emb); acc = ego   (float4 vectorized)
// ---------------------------------------------------------------------------
__global__ void lgcn_init(const float* __restrict__ user_emb,
                          const float* __restrict__ item_emb,
                          float* __restrict__ h,
                          float* __restrict__ acc) {
  const size_t i = (size_t)blockIdx.x * blockDim.x + threadIdx.x;   // float4 idx
  const size_t n4 = NELEMS / 4;
  if (i >= n4) return;
  const size_t userN4 = (size_t)N_USERS * LATENT / 4;
  float4 v = (i < userN4) ? ((const float4*)user_emb)[i]
                          : ((const float4*)item_emb)[i - userN4];
  ((float4*)h)[i]   = v;
  ((float4*)acc)[i] = v;
}

// ---------------------------------------------------------------------------
// zero: y = 0   (float4 vectorized)
// ---------------------------------------------------------------------------
__global__ void lgcn_zero(float* __restrict__ y) {
  const size_t i = (size_t)blockIdx.x * blockDim.x + threadIdx.x;
  const size_t n4 = NELEMS / 4;
  if (i >= n4) return;
  float4 z; z.x = 0.f; z.y = 0.f; z.z = 0.f; z.w = 0.f;
  ((float4*)y)[i] = z;
}

// ---------------------------------------------------------------------------
// COO SpMM: y[row[e], :] += val[e] * x[col[e], :]
// One wave32 per edge. Edge triple is wave-uniform -> readfirstlane forces the
// scalar (s_load) path. Each lane handles a float2 -> coalesced 256B row
// gather (L2-resident: node table is 38.4 MB < 192 MB L2) + 2 f32 atomics.
// ---------------------------------------------------------------------------
__global__ void lgcn_spmm(const int*   __restrict__ row,
                          const int*   __restrict__ col,
                          const float* __restrict__ val,
                          const float* __restrict__ x,
                          float*       __restrict__ y,
                          int nnz) {
  const int lane = threadIdx.x & 31;
  const int wid  = blockIdx.x * (blockDim.x >> 5) + (threadIdx.x >> 5);
  if (wid >= nnz) return;
  const int e = __builtin_amdgcn_readfirstlane(wid);   // SGPR edge index

  const int   r = row[e];        // scalar loads (uniform addr)
  const int   c = col[e];
  const float v = val[e];

  const float2 g = ((const float2*)x)[(size_t)c * (LATENT / 2) + lane];
  float* y0 = y + (size_t)r * LATENT + 2 * lane;
  atomicAdd(y0,     g.x * v);    // no-return global_atomic_add_f32
  atomicAdd(y0 + 1, g.y * v);
}

// ---------------------------------------------------------------------------
// accum: acc += h   (float4 vectorized)
// ---------------------------------------------------------------------------
__global__ void lgcn_accum(float* __restrict__ acc, const float* __restrict__ h) {
  const size_t i = (size_t)blockIdx.x * blockDim.x + threadIdx.x;
  const size_t n4 = NELEMS / 4;
  if (i >= n4) return;
  float4 a = ((const float4*)acc)[i];
  float4 b = ((const float4*)h)[i];
  a.x += b.x; a.y += b.y; a.z += b.z; a.w += b.w;
  ((float4*)acc)[i] = a;
}

// ---------------------------------------------------------------------------
// finalize: acc *= 0.25 via V_WMMA_F32_16X16X4_F32:  D = A_tile x (0.25*I16)
// One wave per 16x16 tile, 4 accumulating WMMAs over K=4 chunks.
//   A (16x4 f32, 2 VGPR): lanes 0-15 -> M=lane, K=0/1; lanes 16-31 -> K=2/3
//   B (4x16 f32, 2 VGPR): VGPR g, half-wave offset 2 -> K = g + 2*half, N=l15
//   C/D (16x16 f32, 8 VGPR): VGPR g: lanes 0-15 -> M=g, lanes 16-31 -> M=g+8
// Grid covers tiles exactly; EXEC all-1s at every WMMA.
// ---------------------------------------------------------------------------
__global__ void lgcn_finalize_wmma(float* __restrict__ acc) {
  const int lane    = threadIdx.x & 31;
  const int tile    = blockIdx.x * (blockDim.x >> 5) + (threadIdx.x >> 5);
  const int colTile = tile & 3;          // 64 / 16 = 4 column tiles
  const int rowTile = tile >> 2;
  const int row0    = rowTile << 4;
  const int c0      = colTile << 4;

  const int half = lane >> 4;            // 0 | 1
  const int l15  = lane & 15;

  // A loads: element (M = l15, K = 2*half + {0,1}) of the current K-chunk.
  const float* abase = acc + (size_t)(row0 + l15) * LATENT + c0 + 2 * half;

  v8f d = {0.f, 0.f, 0.f, 0.f, 0.f, 0.f, 0.f, 0.f};
#pragma unroll
  for (int k = 0; k < 4; ++k) {
    const float2 a2 = *(const float2*)(abase + 4 * k);
    v2f a; a[0] = a2.x; a[1] = a2.y;

    // B chunk = rows [4k, 4k+4) of 0.25*I16
    const int kk = 4 * k + 2 * half;
    v2f b;
    b[0] = (l15 == kk)     ? 0.25f : 0.0f;
    b[1] = (l15 == kk + 1) ? 0.25f : 0.0f;

    d = __builtin_amdgcn_wmma_f32_16x16x4_f32(
            /*neg_a=*/false, a, /*neg_b=*/false, b,
            /*c_mod=*/(short)0, d, /*reuse_a=*/false, /*reuse_b=*/false);
  }

  float* obase = acc + (size_t)(row0 + 8 * half) * LATENT + c0 + l15;
#pragma unroll
  for (int g = 0; g < 8; ++g) {
    obase[(size_t)g * LATENT] = d[g];
  }
}

// ---------------------------------------------------------------------------
extern "C" void kernel_launch(void* const* d_in, const int* in_sizes, int n_in,
                              void* d_out, int out_size, void* d_ws, size_t ws_size,
                              hipStream_t stream) {
  const float* user_emb = (const float*)d_in[0];
  const float* item_emb = (const float*)d_in[1];
  const int*   adj_row  = (const int*)  d_in[2];
  const int*   adj_col  = (const int*)  d_in[3];
  const float* adj_val  = (const float*)d_in[4];
  const int    nnz      = in_sizes[2];

  float* acc = (float*)d_out;            // acc lives directly in d_out
  float* hA  = (float*)d_ws;             // ping-pong h buffers: 2 x 38.4 MB
  float* hB  = hA + NELEMS;

  const int n4 = NELEMS / 4;             // 2,400,000 float4s
  dim3 blkE(256), grdE((n4 + 255) / 256);              // 9375 blocks

  lgcn_init<<<grdE, blkE, 0, stream>>>(user_emb, item_emb, hA, acc);

  const int wavesPerBlock = 8;           // 256 threads = 8 wave32
  dim3 blkS(256), grdS((nnz + wavesPerBlock - 1) / wavesPerBlock);

  float* bufs[2] = {hA, hB};
  for (int layer = 0; layer < NUM_LAYERS; ++layer) {
    float* xin  = bufs[layer & 1];
    float* yout = bufs[(layer + 1) & 1];
    lgcn_zero <<<grdE, blkE, 0, stream>>>(yout);
    lgcn_spmm <<<grdS, blkS, 0, stream>>>(adj_row, adj_col, adj_val, xin, yout, nnz);
    lgcn_accum<<<grdE, blkE, 0, stream>>>(acc, yout);
  }

  // 37500 tiles, 4 waves per 128-thread block -> 9375 blocks, exact cover.
  dim3 blkF(128), grdF((N_NODES / 16) * (LATENT / 16) / 4);
  lgcn_finalize_wmma<<<grdF, blkF, 0, stream>>>(acc);
}